// Affinity_32220844655186
// MI455X (gfx1250) — compile-verified
//
#include <hip/hip_runtime.h>

typedef __attribute__((ext_vector_type(2))) float v2f;
typedef __attribute__((ext_vector_type(8))) float v8f;

#define NB   8          // batch
#define NN   32         // nodes
#define DD   128        // feature dim
#define EE   96         // edges per graph
#define K2   256        // 2*D (GEMM K)
#define OUTN 1024       // NN*NN

// ---- workspace layout (bytes) ----
#define W_OFF       0
#define W_BYTES     (K2*K2*4)                 // 262144, shared weight
#define SAMP_BASE   W_BYTES
#define XT_OFF      0                          // Xt: (96 x 256) row-major
#define YT_OFF      (EE*K2*4)                  // Yt: (96 x 256) row-major
#define P_OFF       (2*EE*K2*4)                // P = Xt*W : (96 x 256)
#define ME_OFF      (3*EE*K2*4)                // Me: (96 x 96)
#define DIAG_OFF    (ME_OFF + EE*EE*4)         // diag: 1024 f32
#define E1SRC_OFF   (DIAG_OFF + OUTN*4)
#define E1TGT_OFF   (E1SRC_OFF + EE*4)
#define E2SRC_OFF   (E1TGT_OFF + EE*4)
#define E2TGT_OFF   (E2SRC_OFF + EE*4)
#define EID1_OFF    (E2TGT_OFF + EE*4)         // edge_id map 32x32 (int, -1 = none)
#define EID2_OFF    (EID1_OFF + OUTN*4)
#define SAMP_STRIDE (EID2_OFF + OUTN*4)        // 345600 B/sample

// W = [[relu(L1+L1^T), relu(L2+L2^T)],[relu(L2+L2^T), relu(L1+L1^T)]] (symmetric)
__global__ void build_weight_k(const float* __restrict__ l1,
                               const float* __restrict__ l2,
                               float* __restrict__ W) {
  int p = blockIdx.x, q = threadIdx.x;
  int pa = p & (DD - 1), qa = q & (DD - 1);
  const float* lam = (((p < DD) == (q < DD)) ? l1 : l2);
  float t = lam[pa * DD + qa] + lam[qa * DD + pa];
  W[p * K2 + q] = fmaxf(t, 0.0f);
}

// Row-major nonzero scan -> edge list + inverse map. One 32-lane wave per (sample, graph).
__global__ void build_edges_k(const float* __restrict__ A_src,
                              const float* __restrict__ A_tgt,
                              char* __restrict__ ws) {
  int s = blockIdx.x >> 1, g = blockIdx.x & 1;
  const float* A = (g ? A_tgt : A_src) + (size_t)s * OUTN;
  char* base = ws + SAMP_BASE + (size_t)s * SAMP_STRIDE;
  int* src = (int*)(base + (g ? E2SRC_OFF : E1SRC_OFF));
  int* tgt = (int*)(base + (g ? E2TGT_OFF : E1TGT_OFF));
  int* eid = (int*)(base + (g ? EID2_OFF : EID1_OFF));
  int i = threadIdx.x;                 // row handled by this lane
  int c = 0;
  for (int j = 0; j < NN; ++j) c += (A[i * NN + j] >= 0.49f) ? 1 : 0;
  __shared__ int cnt[NN];
  cnt[i] = c;
  __syncthreads();
  int e = 0;
  for (int t = 0; t < i; ++t) e += cnt[t];   // exclusive prefix -> row-major order
  for (int j = 0; j < NN; ++j) {
    if (A[i * NN + j] >= 0.49f) {
      src[e] = i; tgt[e] = j; eid[i * NN + j] = e; ++e;
    } else {
      eid[i * NN + j] = -1;
    }
  }
}

// Gather Xt[e,d] = F1[d%128, (d<128?src1:tgt1)[e]] and Yt likewise from F2.
__global__ void build_xy_k(const float* __restrict__ F1,
                           const float* __restrict__ F2,
                           char* __restrict__ ws) {
  int s = blockIdx.y, e = blockIdx.x, d = threadIdx.x;  // d in [0,256)
  char* base = ws + SAMP_BASE + (size_t)s * SAMP_STRIDE;
  const int* s1 = (const int*)(base + E1SRC_OFF);
  const int* t1 = (const int*)(base + E1TGT_OFF);
  const int* s2 = (const int*)(base + E2SRC_OFF);
  const int* t2 = (const int*)(base + E2TGT_OFF);
  const float* f1 = F1 + (size_t)s * DD * NN;
  const float* f2 = F2 + (size_t)s * DD * NN;
  int dd = d & (DD - 1);
  int n1 = (d < DD) ? s1[e] : t1[e];
  int n2 = (d < DD) ? s2[e] : t2[e];
  ((float*)(base + XT_OFF))[e * K2 + d] = f1[dd * NN + n1];
  ((float*)(base + YT_OFF))[e * K2 + d] = f2[dd * NN + n2];
}

// C = A * B^T, A:(16*mt x 256) row-major, B:(16*nt x 256) row-major, K=256.
// One wave per 16x16 tile using V_WMMA_F32_16X16X4_F32 (full fp32 precision).
__global__ void gemm_nt_k256(const float* __restrict__ Abase, long sA,
                             const float* __restrict__ Bbase, long sB,
                             float* __restrict__ Cbase, long sC,
                             int ntiles, int ldc) {
  int s  = blockIdx.y;
  int mt = blockIdx.x / ntiles;
  int nt = blockIdx.x % ntiles;
  const float* A = Abase + (size_t)s * sA;
  const float* B = Bbase + (size_t)s * sB;
  float*       C = Cbase + (size_t)s * sC;
  int lane = threadIdx.x;
  int hi = lane >> 4, lm = lane & 15;
  // A frag: lane holds A[mt*16+lm, k + 2*hi + {0,1}] ; B frag: B[nt*16+lm, k + 2*hi + {0,1}]
  const float* arow = A + (size_t)(mt * 16 + lm) * K2 + 2 * hi;
  const float* brow = B + (size_t)(nt * 16 + lm) * K2 + 2 * hi;
  v8f acc0 = {0.f,0.f,0.f,0.f,0.f,0.f,0.f,0.f};
  v8f acc1 = {0.f,0.f,0.f,0.f,0.f,0.f,0.f,0.f};
#pragma unroll 8
  for (int k = 0; k < K2; k += 8) {
    v2f a0 = *(const v2f*)(arow + k);
    v2f b0 = *(const v2f*)(brow + k);
    acc0 = __builtin_amdgcn_wmma_f32_16x16x4_f32(false, a0, false, b0,
                                                 (short)0, acc0, false, false);
    v2f a1 = *(const v2f*)(arow + k + 4);
    v2f b1 = *(const v2f*)(brow + k + 4);
    acc1 = __builtin_amdgcn_wmma_f32_16x16x4_f32(false, a1, false, b1,
                                                 (short)0, acc1, false, false);
  }
  v8f acc = acc0 + acc1;
#pragma unroll
  for (int v = 0; v < 8; ++v)  // D layout: M = v + 8*hi, N = lm
    C[(size_t)(mt * 16 + v + 8 * hi) * ldc + nt * 16 + lm] = acc[v];
}

// diag[r] = dot(U1[:, r/32], U2[:, r%32])  (Mp = U1^T U2 flattened)
__global__ void diag_k(const float* __restrict__ U1,
                       const float* __restrict__ U2,
                       char* __restrict__ ws) {
  int s = blockIdx.y;
  int r = blockIdx.x * blockDim.x + threadIdx.x;   // 0..1023
  int a = r >> 5, b = r & 31;
  const float* u1 = U1 + (size_t)s * DD * NN;
  const float* u2 = U2 + (size_t)s * DD * NN;
  float acc = 0.f;
#pragma unroll 4
  for (int d = 0; d < DD; ++d) acc += u1[d * NN + a] * u2[d * NN + b];
  ((float*)(ws + SAMP_BASE + (size_t)s * SAMP_STRIDE + DIAG_OFF))[r] = acc;
}

// out[r, c] = Me[e2*96 + e1] if e1,e2 exist else 0 ; +diag on the main diagonal.
// One block per output row; float4 streaming stores (output write is the roofline).
__global__ void scatter_k(const char* __restrict__ ws, float* __restrict__ out) {
  int s = blockIdx.y, r = blockIdx.x;
  int i2 = r >> 5, i1 = r & 31;
  const char* base = ws + SAMP_BASE + (size_t)s * SAMP_STRIDE;
  const int* eid1 = (const int*)(base + EID1_OFF);
  const int* eid2 = (const int*)(base + EID2_OFF);
  const float* Me  = (const float*)(base + ME_OFF);
  const float* dia = (const float*)(base + DIAG_OFF);
  __shared__ int se1[NN], se2[NN];
  int t = threadIdx.x;
  if (t < 32)      se1[t]      = eid1[i1 * NN + t];
  else if (t < 64) se2[t & 31] = eid2[i2 * NN + (t & 31)];
  __syncthreads();
  int c0 = t << 2;                 // 4 consecutive columns share j2
  int e2 = se2[c0 >> 5];
  int j1 = c0 & 31;
  float4 v;
  float* vp = (float*)&v;
#pragma unroll
  for (int k = 0; k < 4; ++k) {
    int e1 = se1[j1 + k];
    vp[k] = (e2 >= 0 && e1 >= 0) ? Me[e2 * EE + e1] : 0.0f;
  }
  if ((unsigned)(r - c0) < 4u) vp[r - c0] += dia[r];
  *(float4*)(out + ((size_t)s * OUTN + r) * OUTN + c0) = v;
}

extern "C" void kernel_launch(void* const* d_in, const int* in_sizes, int n_in,
                              void* d_out, int out_size, void* d_ws, size_t ws_size,
                              hipStream_t stream) {
  const float* A_src = (const float*)d_in[0];
  const float* A_tgt = (const float*)d_in[1];
  const float* F_src = (const float*)d_in[2];
  const float* F_tgt = (const float*)d_in[3];
  const float* U_src = (const float*)d_in[4];
  const float* U_tgt = (const float*)d_in[5];
  const float* lam1  = (const float*)d_in[6];
  const float* lam2  = (const float*)d_in[7];
  char* ws = (char*)d_ws;
  float* W  = (float*)(ws + W_OFF);
  float* Xt = (float*)(ws + SAMP_BASE + XT_OFF);
  float* Yt = (float*)(ws + SAMP_BASE + YT_OFF);
  float* P  = (float*)(ws + SAMP_BASE + P_OFF);
  float* Me = (float*)(ws + SAMP_BASE + ME_OFF);
  const long ss = SAMP_STRIDE / 4;   // per-sample element stride

  build_weight_k<<<dim3(K2), dim3(K2), 0, stream>>>(lam1, lam2, W);
  build_edges_k<<<dim3(2 * NB), dim3(NN), 0, stream>>>(A_src, A_tgt, ws);
  build_xy_k<<<dim3(EE, NB), dim3(K2), 0, stream>>>(F_src, F_tgt, ws);
  // P = Xt * W   (W symmetric -> W == W^T, so NT form is exact): M=96,N=256
  gemm_nt_k256<<<dim3(6 * 16, NB), dim3(32), 0, stream>>>(Xt, ss, W, 0L, P, ss, 16, K2);
  // Me = P * Yt^T : M=96, N=96
  gemm_nt_k256<<<dim3(6 * 6, NB), dim3(32), 0, stream>>>(P, ss, Yt, ss, Me, ss, 6, EE);
  diag_k<<<dim3(4, NB), dim3(256), 0, stream>>>(U_src, U_tgt, ws);
  scatter_k<<<dim3(OUTN, NB), dim3(256), 0, stream>>>(ws, (float*)d_out);
}